// FP16ToFP8Converter_43860206027340
// MI455X (gfx1250) — compile-verified
//
#include <hip/hip_runtime.h>
#include <stdint.h>

typedef uint32_t u32;
typedef unsigned int u32x4 __attribute__((ext_vector_type(4)));
typedef int i32x4 __attribute__((ext_vector_type(4)));
typedef int i32x8 __attribute__((ext_vector_type(8)));

#define THREADS          256
#define ELEMS_PER_BLOCK  512
#define ELEMS_PER_WAVE    64
#define ELEMS_PER_TILE    32   // per-wave sub-tile (double-buffered TDM)
#define LDS_DW_PER_ELEM   20   // 16 data dwords + 4 pad dwords (TDM pad de-swizzle)

#if defined(__HIP_DEVICE_COMPILE__) && defined(__gfx1250__) && \
    __has_builtin(__builtin_amdgcn_tensor_load_to_lds)
#define USE_TDM 1
#else
#define USE_TDM 0
#endif

__device__ __forceinline__ float bitf(u32 b) { return b ? 1.0f : 0.0f; }

// FP16 bit pattern (h) -> 8 FP8-E4M3 pulse floats, exact reference semantics.
__device__ __forceinline__ void convert_and_store(u32 h, float* __restrict__ op) {
  const u32 s = (h >> 15) & 1u;
  const u32 e = (h >> 10) & 31u;
  const u32 m = h & 1023u;

  // --- normal path (e 9..22): RNE-drop 7 mantissa bits
  u32 keep = m >> 7;
  u32 rbit = (m >> 6) & 1u;
  u32 stk  = (m & 63u) ? 1u : 0u;
  u32 mr   = keep + (rbit & (stk | (keep & 1u)));   // 0..8
  u32 ne   = e - 8u + (mr >> 3);
  u32 nm   = mr & 7u;

  // --- subnormal path (e 5..8): RNE((1024+m) >> (16-e)), k clamped [8,11]
  int k = 16 - (int)e;
  k = k < 8 ? 8 : (k > 11 ? 11 : k);
  u32 x     = 1024u + m;
  u32 keep2 = x >> k;
  u32 rbit2 = (x >> (k - 1)) & 1u;
  u32 stk2  = (x & ((1u << (k - 1)) - 1u)) ? 1u : 0u;
  u32 ms    = keep2 + (rbit2 & (stk2 | (keep2 & 1u)));  // 0..8
  u32 se    = ms >> 3;                                   // promote to normal on carry
  u32 sm    = ms & 7u;                                   // 8 -> 0

  u32 oe, om;
  if (e > 22u)      { oe = 15u; om = 6u; }   // overflow -> saturate E4M3 max
  else if (e < 5u)  { oe = 0u;  om = 0u; }   // underflow -> zero
  else if (e <= 8u) { oe = se;  om = sm; }   // subnormal/boundary
  else              { oe = ne;  om = nm; }   // normal

  float4 o0 = make_float4(bitf(s),
                          bitf((oe >> 3) & 1u), bitf((oe >> 2) & 1u), bitf((oe >> 1) & 1u));
  float4 o1 = make_float4(bitf(oe & 1u),
                          bitf((om >> 2) & 1u), bitf((om >> 1) & 1u), bitf(om & 1u));
  *(float4*)(op)     = o0;
  *(float4*)(op + 4) = o1;
}

// Pack 16 pulse floats (0.0f/1.0f) into a 16-bit FP16 pattern, MSB first.
// 1.0f = 0x3F800000 -> (bits>>29)&1 == 1 ; 0.0f -> 0.
__device__ __forceinline__ u32 pack16(const float* p) {
  u32 h = 0;
#pragma unroll
  for (int i = 0; i < 16; i += 4) {
    float4 v = *(const float4*)(p + i);
    h |= ((__float_as_uint(v.x) >> 29) & 1u) << (15 - i);
    h |= ((__float_as_uint(v.y) >> 29) & 1u) << (14 - i);
    h |= ((__float_as_uint(v.z) >> 29) & 1u) << (13 - i);
    h |= ((__float_as_uint(v.w) >> 29) & 1u) << (12 - i);
  }
  return h;
}

#if USE_TDM
// Issue one TDM load: 32 elements = 512 contiguous dwords from `gaddr`,
// written to LDS at `ldsAddr` with 4 pad dwords after every 16 data dwords
// (element pitch 80 B -> 2-way instead of 8-way LDS bank conflicts).
__device__ __forceinline__ void tdm_load_tile(u32 ldsAddr, uint64_t gaddr) {
  u32x4 g0;
  g0[0] = 1u;                                                // count=1 (user D#)
  g0[1] = ldsAddr;                                           // lds_addr (bytes)
  g0[2] = (u32)gaddr;                                        // global_addr[31:0]
  g0[3] = ((u32)(gaddr >> 32) & 0x01FFFFFFu) | 0x80000000u;  // addr[56:32] | type=2

  i32x8 g1;
  g1[0] = 0x06D20000;         // data_size=4B | pad_en | pad_interval=16dw | pad_amount=4dw
  g1[1] = (int)(512u << 16);  // tensor_dim0[15:0] (=512 dwords)
  g1[2] = (int)(1u << 16);    // tensor_dim0 hi=0 | tensor_dim1[15:0]=1
  g1[3] = (int)(512u << 16);  // tensor_dim1 hi=0 | tile_dim0=512
  g1[4] = 1;                  // tile_dim1=1 | tile_dim2=0
  g1[5] = 512;                // tensor_dim0_stride[31:0]
  g1[6] = (int)(512u << 16);  // t0_stride hi=0 | tensor_dim1_stride[15:0]
  g1[7] = 0;                  // tensor_dim1_stride hi
  i32x4 gz = {0, 0, 0, 0};
#if __has_include(<hip/amd_detail/amd_gfx1250_TDM.h>)
  i32x8 gz8 = {0, 0, 0, 0, 0, 0, 0, 0};
  __builtin_amdgcn_tensor_load_to_lds(g0, g1, gz, gz, gz8, 0);
#else
  __builtin_amdgcn_tensor_load_to_lds(g0, g1, gz, gz, 0);
#endif
}
#endif

__global__ __launch_bounds__(THREADS)
void fp16pulse_to_fp8pulse_kernel(const float* __restrict__ in,
                                  float* __restrict__ out, int n) {
  const int blockStart = (int)blockIdx.x * ELEMS_PER_BLOCK;
#if USE_TDM
  __shared__ __align__(16) float tile[ELEMS_PER_BLOCK * LDS_DW_PER_ELEM];  // 40 KB
  if (blockStart + ELEMS_PER_BLOCK <= n) {
    const int lane = (int)threadIdx.x & 31;
    const int wave = (int)threadIdx.x >> 5;
    const int waveElem  = wave * ELEMS_PER_WAVE;              // first element of wave slice
    const u32 ldsBase   = ((u32)(uintptr_t)tile) +
                          (u32)(waveElem * LDS_DW_PER_ELEM * 4);
    const uint64_t gBase =
        (uint64_t)(uintptr_t)(in + (size_t)(blockStart + waveElem) * 16);

    // ---- double-buffered async tensor DMA: issue both sub-tiles up front,
    // overlap tile1's DMA with tile0's compute (per-wave in-order TENSORcnt).
    tdm_load_tile(ldsBase, gBase);
    tdm_load_tile(ldsBase + (u32)(ELEMS_PER_TILE * LDS_DW_PER_ELEM * 4),
                  gBase + (uint64_t)(ELEMS_PER_TILE * 16 * 4));

    __builtin_amdgcn_s_wait_tensorcnt(1);  // tile 0 landed; tile 1 still in flight
    {
      const int el = waveElem + lane;                         // lane-stride-1 elements
      const u32 h = pack16(&tile[el * LDS_DW_PER_ELEM]);
      convert_and_store(h, out + (size_t)(blockStart + el) * 8);
    }
    __builtin_amdgcn_s_wait_tensorcnt(0);  // tile 1 landed
    {
      const int el = waveElem + ELEMS_PER_TILE + lane;
      const u32 h = pack16(&tile[el * LDS_DW_PER_ELEM]);
      convert_and_store(h, out + (size_t)(blockStart + el) * 8);
    }
    return;
  }
#endif
  // Fallback: direct-global vectorized path (tail blocks / non-TDM toolchain).
#pragma unroll
  for (int i = 0; i < ELEMS_PER_BLOCK / THREADS; ++i) {
    const int eg = blockStart + i * THREADS + (int)threadIdx.x;
    if (eg < n) {
      const u32 h = pack16(in + (size_t)eg * 16);
      convert_and_store(h, out + (size_t)eg * 8);
    }
  }
}

extern "C" void kernel_launch(void* const* d_in, const int* in_sizes, int n_in,
                              void* d_out, int out_size, void* d_ws, size_t ws_size,
                              hipStream_t stream) {
  const float* in = (const float*)d_in[0];
  float* out = (float*)d_out;
  const int n = in_sizes[0] / 16;  // elements (fp16 pulses of 16 bits each)
  const int blocks = (n + ELEMS_PER_BLOCK - 1) / ELEMS_PER_BLOCK;
  fp16pulse_to_fp8pulse_kernel<<<dim3(blocks), dim3(THREADS), 0, stream>>>(in, out, n);
}